// GCN_47175920779679
// MI455X (gfx1250) — compile-verified
//
#include <hip/hip_runtime.h>
#include <stdint.h>

#define N_NODES 10000
#define N_EDGES 640000
#define F_IN    128
#define F_HID   128
#define F_OUT   64

typedef __attribute__((ext_vector_type(2))) float v2f;
typedef __attribute__((ext_vector_type(8))) float v8f;
typedef __attribute__((ext_vector_type(4))) uint32_t v4u;
typedef __attribute__((ext_vector_type(8))) int     v8i;
typedef __attribute__((ext_vector_type(4))) int     v4i;

// ---------------------------------------------------------------------------
// Degree / normalization
// ---------------------------------------------------------------------------
__global__ void deg_init_kernel(float* __restrict__ deg, int n) {
    int i = blockIdx.x * blockDim.x + threadIdx.x;
    if (i < n) deg[i] = 1.0f;  // self loop contributes 1 to in-degree
}

__global__ void deg_edges_kernel(const int* __restrict__ dst,
                                 float* __restrict__ deg, int e) {
    int i = blockIdx.x * blockDim.x + threadIdx.x;
    if (i < e) atomicAdd(&deg[dst[i]], 1.0f);
}

__global__ void deg_to_dinv_kernel(float* __restrict__ deg, int n) {
    int i = blockIdx.x * blockDim.x + threadIdx.x;
    if (i < n) deg[i] = rsqrtf(deg[i]);  // deg >= 1 always (self loop)
}

// ---------------------------------------------------------------------------
// TDM: stage a 2-D fp32 tile (nRows x 16 elems, row stride = strideElems)
// from global memory into LDS at byte offset ldsOff.
// D# per CDNA5 ISA 8.3/8.4: group0 {count=1, lds_addr, global_addr, type=2},
// group1 {data_size=4B, tensor_dim0=16, tensor_dim1=nRows, tile_dim0=16,
//          tile_dim1=nRows, tensor_dim0_stride=strideElems}. Groups 2/3 zero
// (2-D tensor). Tracked by TENSORcnt. 6-arg builtin form (this toolchain).
// ---------------------------------------------------------------------------
__device__ __forceinline__ void tdm_load_tile16_f32(uint32_t ldsOff,
                                                    const float* gptr,
                                                    uint32_t nRows,
                                                    uint32_t strideElems) {
    const uint64_t ga = (uint64_t)(uintptr_t)gptr;
    const uint32_t tdim0 = 16u;  // row length (elements) == tile_dim0
    v4u g0;
    g0.x = 1u;                                               // count=1
    g0.y = ldsOff;                                           // lds_addr (bytes)
    g0.z = (uint32_t)ga;                                     // global_addr[31:0]
    g0.w = (uint32_t)((ga >> 32) & 0x1FFFFFFu) | 0x80000000u; // addr[56:32] | type=2
    v8i g1;
    g1[0] = (int)(2u << 16);                                 // data_size = 4 bytes
    g1[1] = (int)((tdim0 & 0xFFFFu) << 16);                  // tensor_dim0 lo16
    g1[2] = (int)(((tdim0 >> 16) & 0xFFFFu) |
                  ((nRows & 0xFFFFu) << 16));                // tdim0 hi | tdim1 lo
    g1[3] = (int)(((nRows >> 16) & 0xFFFFu) |
                  ((tdim0 & 0xFFFFu) << 16));                // tdim1 hi | tile_dim0
    g1[4] = (int)(nRows & 0xFFFFu);                          // tile_dim1 | tile_dim2=0
    g1[5] = (int)strideElems;                                // dim0_stride[31:0]
    g1[6] = 0;                                               // stride0 hi | stride1 lo
    g1[7] = 0;                                               // stride1 hi
    const v4i z4 = {0, 0, 0, 0};
    const v8i z8 = {0, 0, 0, 0, 0, 0, 0, 0};
    __builtin_amdgcn_tensor_load_to_lds(g0, g1, z4, z4, z8, /*cpol=*/0);
}

// ---------------------------------------------------------------------------
// FP32 WMMA GEMM: C[M,N] = A[M,K] * B[K,N]
// Block = 5 wave32s sharing one 16-col N-tile; the K x 16 B tile is staged
// into LDS once per block by the Tensor Data Mover, then consumed with ds
// loads. Each wave owns one 16x16 C tile; K stepped by 4 via
// V_WMMA_F32_16X16X4_F32. Requires (M/16) % 5 == 0 (M=10000 -> 625 = 5*125).
// EXEC stays all-ones everywhere (WMMA requirement).
// ---------------------------------------------------------------------------
#define MT_PER_BLK 5

__global__ __launch_bounds__(32 * MT_PER_BLK)
void wmma_gemm_f32(const float* __restrict__ A, const float* __restrict__ B,
                   float* __restrict__ C, int M, int K, int N) {
    __shared__ float ldsB[128 * 16];                 // up to K=128 rows x 16 cols

    const int lane = threadIdx.x & 31;
    const int wave = threadIdx.x >> 5;
    const int nt   = N >> 4;                         // tiles along N
    const int n0   = (blockIdx.x % nt) << 4;
    const int mg   = blockIdx.x / nt;                // m-group of MT_PER_BLK tiles
    const int m0   = (mg * MT_PER_BLK + wave) << 4;

    // Stage B[k=0..K-1, n0..n0+15] into LDS via TDM (wave 0 only; the branch is
    // wave-uniform and TDM ignores EXEC). Wait TENSORcnt==0 before the barrier.
    if (wave == 0) {
        tdm_load_tile16_f32((uint32_t)(uintptr_t)&ldsB[0], B + n0,
                            (uint32_t)K, (uint32_t)N);
        __builtin_amdgcn_s_wait_tensorcnt(0);
    }
    __syncthreads();

    const int half = lane >> 4;                      // 0: lanes 0-15, 1: 16-31
    const int l16  = lane & 15;

    // A: lane half 0 holds K = k..k+1 of row m0+l16; half 1 holds K = k+2..k+3
    const float* arow = A + (size_t)(m0 + l16) * K + 2 * half;
    // B from LDS: .x row = k+2*half, .y row = k+2*half+1, col = n0+l16
    const float* brow = &ldsB[2 * half * 16 + l16];

    v8f c = {};
    for (int k = 0; k < K; k += 4) {
        v2f a = *(const v2f*)(arow + k);             // contiguous -> global_load_b64
        v2f b;
        b.x = brow[k * 16];
        b.y = brow[k * 16 + 16];
        c = __builtin_amdgcn_wmma_f32_16x16x4_f32(
                /*neg_a=*/false, a, /*neg_b=*/false, b,
                /*c_mod=*/(short)0, c, /*reuse_a=*/false, /*reuse_b=*/false);
    }

    // C: VGPR g -> row m0 + g + 8*half, col n0 + l16
    float* crow = C + (size_t)(m0 + 8 * half) * N + n0 + l16;
#pragma unroll
    for (int g = 0; g < 8; ++g) crow[(size_t)g * N] = c[g];
}

// ---------------------------------------------------------------------------
// agg[i,f] = h[i,f] * dinv[i]^2 + bias[f]   (self-loop term + bias, full init)
// ---------------------------------------------------------------------------
__global__ void self_bias_kernel(const float* __restrict__ h,
                                 const float* __restrict__ dinv,
                                 const float* __restrict__ bias,
                                 float* __restrict__ agg, int n, int F) {
    int idx = blockIdx.x * blockDim.x + threadIdx.x;
    if (idx >= n * F) return;
    int i = idx / F, f = idx - i * F;
    float d = dinv[i];
    agg[idx] = h[idx] * d * d + bias[f];
}

// ---------------------------------------------------------------------------
// Edge scatter: agg[dst] += h[src] * dinv[src]*dinv[dst]
// F/4 threads per edge: float4 gather + 4x global_atomic_add_f32 scatter.
// Working set (~25 MB) is resident in the 192 MB L2 -> L2-atomic bound.
// ---------------------------------------------------------------------------
__global__ void edge_agg_kernel(const int* __restrict__ src,
                                const int* __restrict__ dst,
                                const float* __restrict__ h,
                                const float* __restrict__ dinv,
                                float* __restrict__ agg, int E, int F) {
    int idx = blockIdx.x * blockDim.x + threadIdx.x;
    int cpe = F >> 2;  // float4 chunks per edge
    int e   = idx / cpe;
    if (e >= E) return;
    int c = (idx - e * cpe) << 2;
    int s = src[e], d = dst[e];
    float nrm = dinv[s] * dinv[d];
    const float4 hv = *(const float4*)(h + (size_t)s * F + c);
    float* out = agg + (size_t)d * F + c;
    atomicAdd(out + 0, hv.x * nrm);
    atomicAdd(out + 1, hv.y * nrm);
    atomicAdd(out + 2, hv.z * nrm);
    atomicAdd(out + 3, hv.w * nrm);
}

__global__ void relu_kernel(float* __restrict__ a, int n) {
    int i = blockIdx.x * blockDim.x + threadIdx.x;
    if (i < n) a[i] = fmaxf(a[i], 0.0f);
}

// ---------------------------------------------------------------------------
// Orchestration
// ---------------------------------------------------------------------------
extern "C" void kernel_launch(void* const* d_in, const int* in_sizes, int n_in,
                              void* d_out, int out_size, void* d_ws, size_t ws_size,
                              hipStream_t stream) {
    const float* x   = (const float*)d_in[0];
    const int*   ei  = (const int*)d_in[1];      // [2, E] flat: row0=src, row1=dst
    const float* W1  = (const float*)d_in[2];
    const float* b1  = (const float*)d_in[3];
    const float* W2  = (const float*)d_in[4];
    const float* b2  = (const float*)d_in[5];
    float*       out = (float*)d_out;

    const int* src = ei;
    const int* dst = ei + N_EDGES;

    // Workspace layout
    char*  ws   = (char*)d_ws;
    float* dinv = (float*)ws;                                   //  40 KB
    float* h1   = (float*)(ws + (64 << 10));                    // 5.12 MB
    float* agg1 = (float*)(ws + (64 << 10) + 5242880);          // 5.12 MB
    float* h2   = (float*)(ws + (64 << 10) + 2 * 5242880);      // 2.56 MB

    const int T = 256;

    // Normalization: deg (with self loops) -> dinv = deg^-1/2
    deg_init_kernel<<<(N_NODES + T - 1) / T, T, 0, stream>>>(dinv, N_NODES);
    deg_edges_kernel<<<(N_EDGES + T - 1) / T, T, 0, stream>>>(dst, dinv, N_EDGES);
    deg_to_dinv_kernel<<<(N_NODES + T - 1) / T, T, 0, stream>>>(dinv, N_NODES);

    // Layer 1: h1 = x @ W1 ; agg1 = norm-propagate(h1) + b1 ; relu
    {
        int blocks = ((N_NODES / 16) / MT_PER_BLK) * (F_HID / 16);  // 125*8 = 1000
        wmma_gemm_f32<<<blocks, 32 * MT_PER_BLK, 0, stream>>>(
            x, W1, h1, N_NODES, F_IN, F_HID);
    }
    self_bias_kernel<<<(N_NODES * F_HID + T - 1) / T, T, 0, stream>>>(
        h1, dinv, b1, agg1, N_NODES, F_HID);
    {
        long long work = (long long)N_EDGES * (F_HID / 4);          // 20.48M threads
        edge_agg_kernel<<<(int)((work + T - 1) / T), T, 0, stream>>>(
            src, dst, h1, dinv, agg1, N_EDGES, F_HID);
    }
    relu_kernel<<<(N_NODES * F_HID + T - 1) / T, T, 0, stream>>>(agg1, N_NODES * F_HID);

    // Layer 2: h2 = agg1 @ W2 ; out = norm-propagate(h2) + b2
    {
        int blocks = ((N_NODES / 16) / MT_PER_BLK) * (F_OUT / 16);  // 125*4 = 500
        wmma_gemm_f32<<<blocks, 32 * MT_PER_BLK, 0, stream>>>(
            agg1, W2, h2, N_NODES, F_HID, F_OUT);
    }
    self_bias_kernel<<<(N_NODES * F_OUT + T - 1) / T, T, 0, stream>>>(
        h2, dinv, b2, out, N_NODES, F_OUT);
    {
        long long work = (long long)N_EDGES * (F_OUT / 4);          // 10.24M threads
        edge_agg_kernel<<<(int)((work + T - 1) / T), T, 0, stream>>>(
            src, dst, h2, dinv, out, N_EDGES, F_OUT);
    }
}